// IntraSampleNTXEntLoss_25400436588765
// MI455X (gfx1250) — compile-verified
//
#include <hip/hip_runtime.h>
#include <hip/hip_bf16.h>

typedef float v2f __attribute__((ext_vector_type(2)));
typedef float v8f __attribute__((ext_vector_type(8)));

#define CC 8
#define BB 1024
#define DD 2048
#define TEMP_INV 10.0f   // 1 / 0.1

// D = A*A' accumulate via fp32 WMMA; same fragment used for A and B (Gram trick).
#define GRAM_WMMA(F, ACC) \
  __builtin_amdgcn_wmma_f32_16x16x4_f32(false, (F), false, (F), (short)0, (ACC), false, false)

__global__ __launch_bounds__(128)
void ntxent_gram_kernel(const float* __restrict__ view0,
                        const float* __restrict__ view1,
                        float* __restrict__ per_sample) {
  const int b    = blockIdx.x;        // sample
  const int tid  = threadIdx.x;       // 0..127 (4 waves)
  const int wv   = tid >> 5;          // wave id: owns K slice [wv*512, wv*512+512)
  const int lane = tid & 31;
  const int m    = lane & 15;         // row of X this lane feeds (crop index)

  // X row m: crops 0..7 -> view0, 8..15 -> view1; layout [C, B, D].
  const float* base = (m < 8)
      ? (view0 + ((size_t)m * BB + b) * (size_t)DD)
      : (view1 + ((size_t)(m - 8) * BB + b) * (size_t)DD);

  // A-fragment striping for V_WMMA_F32_16X16X4_F32:
  //   VGPR0: lanes 0-15 -> K=k0,   lanes 16-31 -> K=k0+2
  //   VGPR1: lanes 0-15 -> K=k0+1, lanes 16-31 -> K=k0+3
  // so each lane loads 2 consecutive floats at k0 + 2*(lane>=16).
  const float* rp = base + ((lane >> 4) << 1) + wv * (DD / 4);

  v8f a0 = {}, a1 = {}, a2 = {}, a3 = {};

  // Each wave reduces 512 k-values: 16 iterations x (8 loads + 8 WMMA).
  for (int k = 0; k < DD / 4; k += 32) {
    v2f f0 = *(const v2f*)(rp + k +  0);
    v2f f1 = *(const v2f*)(rp + k +  4);
    v2f f2 = *(const v2f*)(rp + k +  8);
    v2f f3 = *(const v2f*)(rp + k + 12);
    v2f f4 = *(const v2f*)(rp + k + 16);
    v2f f5 = *(const v2f*)(rp + k + 20);
    v2f f6 = *(const v2f*)(rp + k + 24);
    v2f f7 = *(const v2f*)(rp + k + 28);
    a0 = GRAM_WMMA(f0, a0);
    a1 = GRAM_WMMA(f1, a1);
    a2 = GRAM_WMMA(f2, a2);
    a3 = GRAM_WMMA(f3, a3);
    a0 = GRAM_WMMA(f4, a0);
    a1 = GRAM_WMMA(f5, a1);
    a2 = GRAM_WMMA(f6, a2);
    a3 = GRAM_WMMA(f7, a3);
  }
  v8f acc = (a0 + a1) + (a2 + a3);

  // --- spill partial Gram fragments, combine across the 4 waves ---
  __shared__ float Gp[4][16][17];
  {
    const int n  = lane & 15;
    const int mh = lane >> 4;  // C/D layout: VGPR r -> row r + 8*(lane>=16), col lane&15
    #pragma unroll
    for (int r = 0; r < 8; ++r)
      Gp[wv][r + 8 * mh][n] = acc[r];
  }
  __syncthreads();
  for (int idx = tid; idx < 256; idx += 128) {
    const int mm = idx >> 4, nn = idx & 15;
    Gp[0][mm][nn] = (Gp[0][mm][nn] + Gp[1][mm][nn]) + (Gp[2][mm][nn] + Gp[3][mm][nn]);
  }
  __syncthreads();

  // norms from the Gram diagonal (matches F.normalize eps clamp)
  __shared__ float inv[16];
  if (tid < 16) {
    const float nr = __builtin_sqrtf(Gp[0][tid][tid]);
    inv[tid] = 1.0f / fmaxf(nr, 1e-12f);
  }
  __syncthreads();

  // per-anchor NT-Xent term; masked (-1e9) logits contribute exactly 0 -> skip them
  __shared__ float part[8];
  if (tid < 8) {
    const int i = tid;
    const float invi = inv[i];
    float l[16];
    float mx = -__builtin_inff();
    #pragma unroll
    for (int j = 0; j < 16; ++j) {
      const float s = Gp[0][i][j] * invi * inv[j] * TEMP_INV;
      l[j] = (j == i) ? -__builtin_inff() : s;   // exclude self (masked in both views)
      mx = fmaxf(mx, l[j]);
    }
    float se = 0.0f;
    #pragma unroll
    for (int j = 0; j < 16; ++j)
      se += expf(l[j] - mx);                      // exp(-inf)=0
    const float pos = Gp[0][i][i + 8] * invi * inv[i + 8] * TEMP_INV;
    part[i] = (mx + logf(se)) - pos;              // lse - positive logit
  }
  __syncthreads();
  if (tid == 0) {
    float s = 0.0f;
    #pragma unroll
    for (int i = 0; i < 8; ++i) s += part[i];     // fixed order: deterministic
    per_sample[b] = s;
  }
}

// Deterministic fixed-order reduction of 1024 per-sample sums -> mean loss.
__global__ __launch_bounds__(256)
void ntxent_reduce_kernel(const float* __restrict__ per_sample,
                          float* __restrict__ out) {
  __shared__ float s[256];
  const int t = threadIdx.x;
  s[t] = (per_sample[t] + per_sample[t + 256]) +
         (per_sample[t + 512] + per_sample[t + 768]);
  __syncthreads();
  #pragma unroll
  for (int off = 128; off > 0; off >>= 1) {
    if (t < off) s[t] += s[t + off];
    __syncthreads();
  }
  if (t == 0) out[0] = s[0] * (1.0f / (float)(BB * CC));
}

extern "C" void kernel_launch(void* const* d_in, const int* in_sizes, int n_in,
                              void* d_out, int out_size, void* d_ws, size_t ws_size,
                              hipStream_t stream) {
  const float* view0 = (const float*)d_in[0];
  const float* view1 = (const float*)d_in[1];
  float* per_sample  = (float*)d_ws;     // 1024 floats of scratch
  float* out         = (float*)d_out;

  ntxent_gram_kernel<<<BB, 128, 0, stream>>>(view0, view1, per_sample);
  ntxent_reduce_kernel<<<1, 256, 0, stream>>>(per_sample, out);
}